// PixelShuffle_13666585936261
// MI455X (gfx1250) — compile-verified
//
#include <hip/hip_runtime.h>
#include <stdint.h>

// PixelUnshuffle(s=2) + 2x2 avg-pool, fused. Pure streaming: ~243 MB total
// traffic, 0.035 FLOP/byte -> HBM-bound (~10.4 us at 23.3 TB/s).
// CDNA5 path: async global->LDS b128 loads (ASYNCcnt), LDS remap,
// async LDS->global b128 stores. All transactions aligned 16B.
//
// Per block (256 threads): one (b, ho) full output row, 256 output pixels.
//   inputs : images rows 2ho,2ho+1 (512 px * 3 ch = 6144 B each)
//            maps   rows 2ho,2ho+1 (512 px       = 2048 B each)
//   output : 256 px * 13 ch * 4 B = 13312 B contiguous, 16B-aligned.
//
// LDS layout (float index / byte offset):
//   [   0,1536)  @     0  img row a=0
//   [1536,3072)  @  6144  img row a=1
//   [3072,3584)  @ 12288  map row a=0
//   [3584,4096)  @ 14336  map row a=1
//   [4096,7424)  @ 16384  staged output (3328 floats)

__global__ __launch_bounds__(256) void pixelshuffle_pool_kernel(
    const float* __restrict__ images,
    const float* __restrict__ maps,
    float* __restrict__ out) {
  __shared__ __align__(16) float lds[7424];

  const int t   = threadIdx.x;
  const int bid = blockIdx.x;
  const int ho  = bid & 255;
  const int b   = bid >> 8;

  // Low 32 bits of the generic pointer to an LDS object = LDS byte offset.
  const uint32_t lds_base = (uint32_t)(uintptr_t)(&lds[0]);

  // ---------------- Phase 1: async global -> LDS (1024 x b128, 4/lane) -------
  // chunks q in [0,768): image rows (384 float4 per row)
  // chunks q in [768,1024): map rows (128 float4 per row)
#pragma unroll
  for (int i = 0; i < 4; ++i) {
    const int q = t + i * 256;
    uint64_t g;
    uint32_t l;
    if (q < 768) {
      const int a     = q / 384;       // input row within the 2x2 (H offset)
      const int chunk = q - a * 384;   // float4 index within 6144 B row slab
      const size_t row = (size_t)(b * 512 + 2 * ho + a);
      const float* src = images + row * 1536 + (size_t)chunk * 4;
      g = (uint64_t)(uintptr_t)src;
      l = lds_base + (uint32_t)(a * 6144 + chunk * 16);
    } else {
      const int qq    = q - 768;
      const int a     = qq >> 7;
      const int chunk = qq & 127;
      const size_t row = (size_t)(b * 512 + 2 * ho + a);
      const float* src = maps + row * 512 + (size_t)chunk * 4;
      g = (uint64_t)(uintptr_t)src;
      l = lds_base + (uint32_t)(12288 + a * 2048 + chunk * 16);
    }
    asm volatile("global_load_async_to_lds_b128 %0, %1, off"
                 :: "v"(l), "v"(g)
                 : "memory");
  }

#if defined(__has_builtin) && __has_builtin(__builtin_amdgcn_s_wait_asynccnt)
  __builtin_amdgcn_s_wait_asynccnt(0);
#else
  asm volatile("s_wait_asynccnt 0" ::: "memory");
#endif
  __syncthreads();

  // ---------------- Phase 2: remap, one output pixel per thread --------------
  // out[k] with k = c*4 + bb*2 + a  <-  imgrow_a[(2p+bb)*3 + c]
  // out[12] = 0.25 * (2x2 maps block)
  {
    const int p = t;
    float o[13];
#pragma unroll
    for (int c = 0; c < 3; ++c) {
#pragma unroll
      for (int bb = 0; bb < 2; ++bb) {
        o[c * 4 + bb * 2 + 0] = lds[0    + 6 * p + bb * 3 + c];
        o[c * 4 + bb * 2 + 1] = lds[1536 + 6 * p + bb * 3 + c];
      }
    }
    o[12] = 0.25f * (lds[3072 + 2 * p] + lds[3072 + 2 * p + 1] +
                     lds[3584 + 2 * p] + lds[3584 + 2 * p + 1]);
#pragma unroll
    for (int k = 0; k < 13; ++k) {
      lds[4096 + 13 * p + k] = o[k];
    }
  }
  __syncthreads();

  // ---------------- Phase 3: async LDS -> global (832 x b128) ----------------
  const size_t obase = (size_t)(b * 256 + ho) * 3328;  // floats
#pragma unroll
  for (int i = 0; i < 3; ++i) {
    const int q = t + i * 256;
    const float* dstp = out + obase + (size_t)q * 4;
    const uint64_t go = (uint64_t)(uintptr_t)dstp;
    const uint32_t lo = lds_base + 16384u + (uint32_t)q * 16u;
    asm volatile("global_store_async_from_lds_b128 %0, %1, off"
                 :: "v"(go), "v"(lo)
                 : "memory");
  }
  if (t < 64) {
    const int q = 768 + t;
    const float* dstp = out + obase + (size_t)q * 4;
    const uint64_t go = (uint64_t)(uintptr_t)dstp;
    const uint32_t lo = lds_base + 16384u + (uint32_t)q * 16u;
    asm volatile("global_store_async_from_lds_b128 %0, %1, off"
                 :: "v"(go), "v"(lo)
                 : "memory");
  }
  // Implicit S_WAIT_IDLE at S_ENDPGM drains the outstanding async stores.
}

extern "C" void kernel_launch(void* const* d_in, const int* in_sizes, int n_in,
                              void* d_out, int out_size, void* d_ws, size_t ws_size,
                              hipStream_t stream) {
  const float* images = (const float*)d_in[0];
  const float* maps   = (const float*)d_in[1];
  float* out          = (float*)d_out;

  // 32 batches * 256 output rows; one full row per block
  const int nblk = 32 * 256;
  pixelshuffle_pool_kernel<<<nblk, 256, 0, stream>>>(images, maps, out);
}